// InstrumentsChecker_1279900254760
// MI455X (gfx1250) — compile-verified
//
#include <hip/hip_runtime.h>
#include <stdint.h>

// Problem constants (match reference)
#define B_DIM    64
#define S_DIM    4096
#define N_INSTR  128
#define NPOS     (B_DIM * S_DIM)          // 262144 positions
#define TILE_POS 32                        // positions per TDM tile
#define TILE_FLOATS (TILE_POS * N_INSTR)   // 4096 f32 = 16 KB
#define TILE_BYTES  (TILE_FLOATS * 4)
#define WAVES_PER_BLOCK 8
#define BLOCK_THREADS   256
#define NTILES   (NPOS / TILE_POS)         // 8192
#define TILES_PER_BLOCK 4
#define PASS1_BLOCKS (NTILES / TILES_PER_BLOCK)  // 2048

typedef __attribute__((ext_vector_type(4))) unsigned int v4u;
typedef __attribute__((ext_vector_type(8))) unsigned int v8u;

// -----------------------------------------------------------------------------
// Tensor Data Mover: async-load one contiguous 16KB tile of f32 scores into LDS.
// D# built per CDNA5 ISA ch.8 (group0: count/lds_addr/global_addr/type=2;
// group1: data_size=4B, 1 x TILE_FLOATS tile). Inline asm (portable across
// ROCm7.2 / amdgpu-toolchain builtin arity mismatch). Tracked by TENSORcnt.
// -----------------------------------------------------------------------------
__device__ __forceinline__ void tdm_load_tile(const float* gsrc, unsigned lds_off) {
  uint64_t ga = (uint64_t)(uintptr_t)gsrc;
  v4u g0;
  g0[0] = 1u;                                   // count=1, user descriptor
  g0[1] = lds_off;                              // LDS byte address
  g0[2] = (unsigned)ga;                         // global_addr[31:0]
  g0[3] = (unsigned)((ga >> 32) & 0x01FFFFFFu)  // global_addr[56:32]
        | (2u << 30);                           // type=2 ("image")
  v8u g1;
  g1[0] = (2u << 16);                           // workgroup_mask=0, data_size=4B
  g1[1] = ((unsigned)TILE_FLOATS & 0xFFFFu) << 16;  // tensor_dim0[15:0]
  g1[2] = (1u << 16) | (((unsigned)TILE_FLOATS >> 16) & 0xFFFFu); // tensor_dim1=1
  g1[3] = ((unsigned)TILE_FLOATS & 0xFFFFu) << 16;  // tile_dim0 = TILE_FLOATS
  g1[4] = 1u;                                   // tile_dim1 = 1, tile_dim2 = 0
  g1[5] = (unsigned)TILE_FLOATS;                // tensor_dim0_stride (low 32)
  g1[6] = 0u;
  g1[7] = 0u;
  asm volatile("tensor_load_to_lds %0, %1" :: "s"(g0), "s"(g1) : "memory");
}

// -----------------------------------------------------------------------------
// Pass 1: argmax over 128 instruments per position, TDM double-buffered LDS.
// One wave per position per step: lane l owns elements 4l..4l+3, then a
// 5-step wave32 butterfly with first-index tie-break (== jnp.argmax).
// -----------------------------------------------------------------------------
__global__ __launch_bounds__(BLOCK_THREADS)
void pass1_argmax(const float* __restrict__ scores,
                  const int* __restrict__ types,
                  unsigned char* __restrict__ ws_am,
                  unsigned* __restrict__ ws_present) {
  __shared__ __align__(16) float tile[2][TILE_FLOATS];
  const unsigned lane = threadIdx.x & 31u;
  const unsigned wave = threadIdx.x >> 5;
  const unsigned ldsBase = (unsigned)(uintptr_t)&tile[0][0];

  const int firstTile = blockIdx.x * TILES_PER_BLOCK;

  if (wave == 0) {
    tdm_load_tile(scores + (size_t)firstTile * TILE_FLOATS, ldsBase);
  }

  for (int t = 0; t < TILES_PER_BLOCK; ++t) {
    const int cur = t & 1;
    if (wave == 0) {
      __builtin_amdgcn_s_wait_tensorcnt(0);     // s_wait_tensorcnt 0
    }
    __syncthreads();                            // tile[cur] visible to all waves
    if (wave == 0 && (t + 1) < TILES_PER_BLOCK) {
      tdm_load_tile(scores + (size_t)(firstTile + t + 1) * TILE_FLOATS,
                    ldsBase + (unsigned)((1 - cur) * TILE_BYTES));
    }

    const int tileIdx = firstTile + t;
    for (int p = (int)wave; p < TILE_POS; p += WAVES_PER_BLOCK) {
      const float4* row = (const float4*)&tile[cur][p * N_INSTR];
      float4 v = row[lane];                     // ds_load_b128
      float best = v.x;
      int   bidx = (int)(lane * 4u);
      if (v.y > best) { best = v.y; bidx = (int)(lane * 4u) + 1; }
      if (v.z > best) { best = v.z; bidx = (int)(lane * 4u) + 2; }
      if (v.w > best) { best = v.w; bidx = (int)(lane * 4u) + 3; }
#pragma unroll
      for (int off = 16; off > 0; off >>= 1) {
        float ov = __shfl_xor(best, off, 32);
        int   oi = __shfl_xor(bidx, off, 32);
        if (ov > best || (ov == best && oi < bidx)) { best = ov; bidx = oi; }
      }
      if (lane == 0) {
        const int pos = tileIdx * TILE_POS + p;
        ws_am[pos] = (unsigned char)bidx;
        if (types[pos] == 1) {
          const int b = pos >> 12;              // / S_DIM
          atomicOr(&ws_present[b * 4 + (bidx >> 5)], 1u << (bidx & 31));
        }
      }
    }
    __syncthreads();                            // done reading tile[cur]
  }
}

// -----------------------------------------------------------------------------
// Pass 2: total += #(type==1) + #(type==3 & instrument not defined in batch)
// -----------------------------------------------------------------------------
__global__ __launch_bounds__(256)
void pass2_count(const int* __restrict__ types,
                 const unsigned char* __restrict__ ws_am,
                 const unsigned* __restrict__ ws_present,
                 int* __restrict__ out) {
  const int stride = (int)(gridDim.x * blockDim.x);
  int local = 0;
  for (int pos = (int)(blockIdx.x * blockDim.x + threadIdx.x); pos < NPOS; pos += stride) {
    const int ty = types[pos];
    if (ty == 1) {
      local += 1;
    } else if (ty == 3) {
      const int am = (int)ws_am[pos];
      const int b  = pos >> 12;
      const unsigned w = ws_present[b * 4 + (am >> 5)];
      if (((w >> (am & 31)) & 1u) == 0u) local += 1;
    }
  }
#pragma unroll
  for (int off = 16; off > 0; off >>= 1) local += __shfl_xor(local, off, 32);
  __shared__ int warpSum[WAVES_PER_BLOCK];
  if ((threadIdx.x & 31u) == 0u) warpSum[threadIdx.x >> 5] = local;
  __syncthreads();
  if (threadIdx.x == 0) {
    int s = 0;
#pragma unroll
    for (int i = 0; i < WAVES_PER_BLOCK; ++i) s += warpSum[i];
    atomicAdd(out, s);
  }
}

// -----------------------------------------------------------------------------
// Pass 3: total -= sum_b popcount(present[b])  (the |unique| term of reg1)
// -----------------------------------------------------------------------------
__global__ __launch_bounds__(256)
void pass3_popc(const unsigned* __restrict__ ws_present, int* __restrict__ out) {
  int c = __popc(ws_present[threadIdx.x]);      // 256 words, 1 block of 256
#pragma unroll
  for (int off = 16; off > 0; off >>= 1) c += __shfl_xor(c, off, 32);
  __shared__ int warpSum[WAVES_PER_BLOCK];
  if ((threadIdx.x & 31u) == 0u) warpSum[threadIdx.x >> 5] = c;
  __syncthreads();
  if (threadIdx.x == 0) {
    int s = 0;
#pragma unroll
    for (int i = 0; i < WAVES_PER_BLOCK; ++i) s += warpSum[i];
    atomicAdd(out, -s);
  }
}

__global__ void init_kernel(unsigned* __restrict__ ws_present, int* __restrict__ out) {
  if (threadIdx.x < 256) ws_present[threadIdx.x] = 0u;
  if (threadIdx.x == 0) out[0] = 0;
}

extern "C" void kernel_launch(void* const* d_in, const int* in_sizes, int n_in,
                              void* d_out, int out_size, void* d_ws, size_t ws_size,
                              hipStream_t stream) {
  const int*   types  = (const int*)d_in[0];     // max_pred_types [B,S] int32
  const float* scores = (const float*)d_in[1];   // instrument_scores [B,S,128] f32
  int* out = (int*)d_out;                        // single int32 result

  // Scratch layout: [0,1KB) present bitmasks (64 batches x 4 u32),
  //                 [1KB, 1KB+256KB) am per position (u8).
  unsigned*      ws_present = (unsigned*)d_ws;
  unsigned char* ws_am      = (unsigned char*)d_ws + 1024;

  init_kernel<<<1, 256, 0, stream>>>(ws_present, out);
  pass1_argmax<<<PASS1_BLOCKS, BLOCK_THREADS, 0, stream>>>(scores, types, ws_am, ws_present);
  pass2_count<<<512, 256, 0, stream>>>(types, ws_am, ws_present, out);
  pass3_popc<<<1, 256, 0, stream>>>(ws_present, out);
}